// InteractionEncoder_60790967107676
// MI455X (gfx1250) — compile-verified
//
#include <hip/hip_runtime.h>

typedef __attribute__((ext_vector_type(2))) float v2f;
typedef __attribute__((ext_vector_type(8))) float v8f;

#define D_DIM 256
#define C_DIM 1024
#define NREL  13
#define NBLOCKS_ACC 512

// ---------------------------------------------------------------------------
// Kernel A: S = character_slots (1024x256) @ W1_top (256x256), FP32 WMMA.
// One wave (32 threads) per 16x16 output tile; K-loop of 64 x (16x16x4).
// grid = (1024/16, 256/16) = (64, 16)
// ---------------------------------------------------------------------------
__global__ __launch_bounds__(32) void ie_precompute_S(
    const float* __restrict__ slots,   // (1024, 256) row-major
    const float* __restrict__ W1,      // (512, 256) row-major; top half used
    float* __restrict__ S)             // (1024, 256) row-major
{
    const int m0   = blockIdx.x * 16;
    const int n0   = blockIdx.y * 16;
    const int lane = threadIdx.x;      // 0..31 (wave32)
    const int hi   = lane >> 4;        // 0: lanes 0-15, 1: lanes 16-31
    const int l    = lane & 15;

    v8f c = {};
    for (int kb = 0; kb < D_DIM; kb += 4) {
        const int ka = kb + 2 * hi;
        // A tile 16x4: lane holds A[m=l][ka], A[m=l][ka+1] (contiguous in k)
        v2f a = *(const v2f*)(slots + (m0 + l) * D_DIM + ka);
        // B tile 4x16: lane holds B[ka][n0+l], B[ka+1][n0+l]
        v2f b;
        b.x = W1[(ka + 0) * D_DIM + n0 + l];
        b.y = W1[(ka + 1) * D_DIM + n0 + l];
        // (neg_a, A, neg_b, B, c_mod, C, reuse_a, reuse_b)
        c = __builtin_amdgcn_wmma_f32_16x16x4_f32(
                false, a, false, b, (short)0, c, false, false);
    }
    // D layout: VGPR j holds row M = j + 8*hi, N = l
#pragma unroll
    for (int j = 0; j < 8; ++j)
        S[(m0 + j + 8 * hi) * D_DIM + n0 + l] = c[j];
}

// ---------------------------------------------------------------------------
// Kernel R: R'[r] = rel_embed[r] @ (W1_top + W1_bot) + b1   (13 x 256)
// grid = (13), block = 256. rel_embed reads are wave-uniform -> s_load.
// ---------------------------------------------------------------------------
__global__ __launch_bounds__(256) void ie_precompute_R(
    const float* __restrict__ rel_embed,  // (13, 256)
    const float* __restrict__ W1,         // (512, 256)
    const float* __restrict__ b1,         // (256)
    float* __restrict__ Rp)               // (13, 256)
{
    const int r = blockIdx.x;
    const int d = threadIdx.x;
    float acc = b1[d];
    for (int k = 0; k < D_DIM; ++k) {
        const float re = rel_embed[r * D_DIM + k];          // uniform
        acc += re * (W1[k * D_DIM + d] + W1[(D_DIM + k) * D_DIM + d]);
    }
    Rp[r * D_DIM + d] = acc;
}

// ---------------------------------------------------------------------------
// Kernel B (hot loop): per-block partial of Σ silu(S[a]+S[b]+R'[rel]).
// 256 threads = one thread per hidden dim; index loads are block-uniform
// (scalar loads); table gathers are 1KB coalesced rows resident in L2.
// ---------------------------------------------------------------------------
__global__ __launch_bounds__(256) void ie_accumulate(
    const float* __restrict__ S,        // (1024, 256), L2-resident (1 MB)
    const float* __restrict__ Rp,       // (13, 256)
    const int*   __restrict__ idx_a,
    const int*   __restrict__ idx_b,
    const int*   __restrict__ rel_ids,
    float* __restrict__ partials,       // (NBLOCKS_ACC, 256)
    int n_total, int per_block)
{
    const int t    = threadIdx.x;
    const int base = blockIdx.x * per_block;
    float acc = 0.0f;

#pragma unroll 4
    for (int i = 0; i < per_block; ++i) {
        const int n = base + i;
        if (n >= n_total) break;
        // Block-uniform index fetches -> scalar loads through K$
        const int a = idx_a[n];
        const int b = idx_b[n];
        const int r = rel_ids[n];
        const float z = S[a * D_DIM + t] + S[b * D_DIM + t] + Rp[r * D_DIM + t];
        // silu(z) = z * sigmoid(z) = z * rcp(1 + exp(-z)) : v_exp_f32 + v_rcp_f32
        acc += z * __builtin_amdgcn_rcpf(1.0f + __expf(-z));
    }
    partials[blockIdx.x * D_DIM + t] = acc;
}

// ---------------------------------------------------------------------------
// Kernel C: deterministic reduction of partials -> H (256), then
// out = H @ W2 + N * b2. Single block => fixed summation order.
// ---------------------------------------------------------------------------
__global__ __launch_bounds__(256) void ie_finalize(
    const float* __restrict__ partials, // (NBLOCKS_ACC, 256)
    const float* __restrict__ W2,       // (256, 256)
    const float* __restrict__ b2,       // (256)
    float* __restrict__ out,            // (256)
    int nblocks, float n_as_float)
{
    __shared__ float Hs[D_DIM];
    const int t = threadIdx.x;

    float h = 0.0f;
    for (int b = 0; b < nblocks; ++b)
        h += partials[b * D_DIM + t];
    Hs[t] = h;
    __syncthreads();

    float acc = n_as_float * b2[t];
    for (int k = 0; k < D_DIM; ++k)
        acc += Hs[k] * W2[k * D_DIM + t];   // coalesced W2 column walk
    out[t] = acc;
}

// ---------------------------------------------------------------------------
extern "C" void kernel_launch(void* const* d_in, const int* in_sizes, int n_in,
                              void* d_out, int out_size, void* d_ws, size_t ws_size,
                              hipStream_t stream) {
    const float* slots     = (const float*)d_in[0];
    const int*   idx_a     = (const int*)  d_in[1];
    const int*   idx_b     = (const int*)  d_in[2];
    const int*   rel_ids   = (const int*)  d_in[3];
    const float* rel_embed = (const float*)d_in[4];
    const float* W1        = (const float*)d_in[5];
    const float* b1        = (const float*)d_in[6];
    const float* W2        = (const float*)d_in[7];
    const float* b2        = (const float*)d_in[8];
    float*       out       = (float*)d_out;
    const int    N         = in_sizes[1];

    // Workspace layout
    char*  ws       = (char*)d_ws;
    float* S        = (float*)(ws);                                   // 1 MB
    float* Rp       = (float*)(ws + (size_t)C_DIM * D_DIM * 4);       // 13 KB (pad 16K)
    float* partials = (float*)(ws + (size_t)C_DIM * D_DIM * 4 + 16384);

    const int per_block = (N + NBLOCKS_ACC - 1) / NBLOCKS_ACC;        // 1024

    dim3 gridS(C_DIM / 16, D_DIM / 16);  // (64, 16)
    ie_precompute_S<<<gridS, 32, 0, stream>>>(slots, W1, S);
    ie_precompute_R<<<NREL, D_DIM, 0, stream>>>(rel_embed, W1, b1, Rp);
    ie_accumulate<<<NBLOCKS_ACC, D_DIM, 0, stream>>>(S, Rp, idx_a, idx_b, rel_ids,
                                                     partials, N, per_block);
    ie_finalize<<<1, D_DIM, 0, stream>>>(partials, W2, b2, out,
                                         NBLOCKS_ACC, (float)N);
}